// MusicTransformer_56590489092572
// MI455X (gfx1250) — compile-verified
//
#include <hip/hip_runtime.h>
#include <math.h>

// ---------------- problem constants (from the reference) ----------------
#define S_      1024
#define DIN_    128
#define DM_     512          // d_model
#define NH_     8
#define HD_     64
#define NL_     4
#define DOUT_   390
#define DFF_    2048
#define ER_ROWS 2047         // 2*MAXS-1
#define G_LD    2048         // padded leading dim for the skew buffer
#define EPS_    1e-5f

typedef __attribute__((ext_vector_type(2))) float v2f;
typedef __attribute__((ext_vector_type(8))) float v8f;

#define ACT_NONE 0
#define ACT_GELU 1

__device__ __forceinline__ float gelu_exact(float x) {
    // gelu(x) = 0.5 x (1 + erf(x/sqrt(2)))  (approximate=False)
    return 0.5f * x * (1.0f + erff(x * 0.70710678118654752f));
}

__device__ __forceinline__ v8f wmma_f32(v2f a, v2f b, v8f c) {
    return __builtin_amdgcn_wmma_f32_16x16x4_f32(
        false, a, false, b, (short)0, c, false, false);
}

// ======================================================================
// C[M,N] = act(A[M,K] @ B[K,N] + bias) (+ residual)
// 128 threads = 4 wave32; each wave computes a 16x32 strip (two 16x16
// WMMA tiles sharing the A fragment). V_WMMA_F32_16X16X4_F32 layouts:
//   A 16x4 : lane m=lane&15 -> row m; kb=2*(lane>>4) selects the K pair.
//   B 4x16 : lane n=lane&15 -> col n, same kb pairing.
//   C 16x16: VGPR r -> row r (lanes 0-15) / r+8 (lanes 16-31), col=lane&15.
// GUARD=false: branch-free inner loop (N multiple of 32).
// GUARD=true : clamped addresses + value selects (no exec divergence).
// ======================================================================
template<bool GUARD>
__global__ __launch_bounds__(128)
void gemm_nn_kernel(const float* __restrict__ A, int lda,
                    const float* __restrict__ B, int ldb,
                    const float* __restrict__ bias,
                    const float* __restrict__ Rsd, int ldr,
                    float* __restrict__ C, int ldc,
                    int N, int K, int act)
{
    const int lane = threadIdx.x & 31;
    const int wave = threadIdx.x >> 5;
    const int m    = lane & 15;
    const int kb   = (lane >> 4) << 1;                 // 0 or 2
    const int row0 = blockIdx.y << 4;
    const int col0 = ((blockIdx.x << 2) + wave) << 5;  // 32 cols per wave
    if (col0 >= N) return;                             // wave-uniform

    const int  c0  = col0 + m;
    const int  c1  = col0 + 16 + m;
    const bool ok0 = GUARD ? (c0 < N) : true;
    const bool ok1 = GUARD ? (c1 < N) : true;
    const int  c0c = GUARD ? (ok0 ? c0 : N - 1) : c0;  // clamped (always valid)
    const int  c1c = GUARD ? (ok1 ? c1 : N - 1) : c1;

    v8f acc0 = {}, acc1 = {};
    const float* arow = A + (size_t)(row0 + m) * lda + kb;
    // B row pointers for rows (kb) and (kb+1); advance by 4 rows per step.
    const float* b0lo = B + (size_t)kb * ldb + c0c;
    const float* b0hi = b0lo + ldb;
    const float* b1lo = B + (size_t)kb * ldb + c1c;
    const float* b1hi = b1lo + ldb;
    const size_t step = (size_t)ldb << 2;              // 4 rows

    for (int k = 0; k < K; k += 4) {
        v2f a, b0, b1;
        a.x = arow[k];
        a.y = arow[k + 1];
        float t00 = b0lo[0], t01 = b0hi[0];
        float t10 = b1lo[0], t11 = b1hi[0];
        if (GUARD) {
            t00 = ok0 ? t00 : 0.0f;  t01 = ok0 ? t01 : 0.0f;
            t10 = ok1 ? t10 : 0.0f;  t11 = ok1 ? t11 : 0.0f;
        }
        b0.x = t00; b0.y = t01;
        b1.x = t10; b1.y = t11;
        acc0 = wmma_f32(a, b0, acc0);
        acc1 = wmma_f32(a, b1, acc1);
        b0lo += step; b0hi += step; b1lo += step; b1hi += step;
    }

    const int rbase = row0 + ((lane >> 4) << 3);
    if (ok0) {
        const float bv = bias ? bias[c0] : 0.0f;
#pragma unroll
        for (int r = 0; r < 8; ++r) {
            const int row = rbase + r;
            float x = acc0[r] + bv;
            if (act == ACT_GELU) x = gelu_exact(x);
            if (Rsd) x += Rsd[(size_t)row * ldr + c0];
            C[(size_t)row * ldc + c0] = x;
        }
    }
    if (ok1) {
        const float bv = bias ? bias[c1] : 0.0f;
#pragma unroll
        for (int r = 0; r < 8; ++r) {
            const int row = rbase + r;
            float x = acc1[r] + bv;
            if (act == ACT_GELU) x = gelu_exact(x);
            if (Rsd) x += Rsd[(size_t)row * ldr + c1];
            C[(size_t)row * ldc + c1] = x;
        }
    }
}

// ======================================================================
// C[M,N] = alpha * A[M,K] @ B[N,K]^T   (both row-major, strided)
// Used for Q·K^T (scores) and Q·Er^T (skew buffer). Same tiling as above;
// here each lane's B fragment is 2 consecutive floats -> b64 loads.
// ======================================================================
template<bool GUARD>
__global__ __launch_bounds__(128)
void gemm_nt_kernel(const float* __restrict__ A, int lda,
                    const float* __restrict__ B, int ldb,
                    float* __restrict__ C, int ldc,
                    int N, int K, float alpha)
{
    const int lane = threadIdx.x & 31;
    const int wave = threadIdx.x >> 5;
    const int m    = lane & 15;
    const int kb   = (lane >> 4) << 1;
    const int row0 = blockIdx.y << 4;
    const int col0 = ((blockIdx.x << 2) + wave) << 5;
    if (col0 >= N) return;

    const int  c0  = col0 + m;
    const int  c1  = col0 + 16 + m;
    const bool ok0 = GUARD ? (c0 < N) : true;
    const bool ok1 = GUARD ? (c1 < N) : true;
    const int  c0c = GUARD ? (ok0 ? c0 : N - 1) : c0;
    const int  c1c = GUARD ? (ok1 ? c1 : N - 1) : c1;

    v8f acc0 = {}, acc1 = {};
    const float* arow = A + (size_t)(row0 + m) * lda + kb;
    const float* br0  = B + (size_t)c0c * ldb + kb;
    const float* br1  = B + (size_t)c1c * ldb + kb;

    for (int k = 0; k < K; k += 4) {
        v2f a, b0, b1;
        a.x = arow[k];
        a.y = arow[k + 1];
        float t00 = br0[k], t01 = br0[k + 1];
        float t10 = br1[k], t11 = br1[k + 1];
        if (GUARD) {
            t00 = ok0 ? t00 : 0.0f;  t01 = ok0 ? t01 : 0.0f;
            t10 = ok1 ? t10 : 0.0f;  t11 = ok1 ? t11 : 0.0f;
        }
        b0.x = t00; b0.y = t01;
        b1.x = t10; b1.y = t11;
        acc0 = wmma_f32(a, b0, acc0);
        acc1 = wmma_f32(a, b1, acc1);
    }

    const int rbase = row0 + ((lane >> 4) << 3);
    if (ok0) {
#pragma unroll
        for (int r = 0; r < 8; ++r)
            C[(size_t)(rbase + r) * ldc + c0] = alpha * acc0[r];
    }
    if (ok1) {
#pragma unroll
        for (int r = 0; r < 8; ++r)
            C[(size_t)(rbase + r) * ldc + c1] = alpha * acc1[r];
    }
}

// ======================================================================
// Row-wise LayerNorm over D elements; one 128-thread block per row.
// ======================================================================
__global__ __launch_bounds__(128)
void layernorm_kernel(const float* __restrict__ x,
                      const float* __restrict__ g,
                      const float* __restrict__ b,
                      float* __restrict__ y, int D)
{
    const int row = blockIdx.x;
    const float* xr = x + (size_t)row * D;
    float*       yr = y + (size_t)row * D;
    __shared__ float s1[128], s2[128];

    float lsum = 0.0f, lsq = 0.0f;
    for (int i = threadIdx.x; i < D; i += 128) {
        const float v = xr[i];
        lsum += v; lsq += v * v;
    }
    s1[threadIdx.x] = lsum; s2[threadIdx.x] = lsq;
    __syncthreads();
    for (int off = 64; off > 0; off >>= 1) {
        if (threadIdx.x < off) {
            s1[threadIdx.x] += s1[threadIdx.x + off];
            s2[threadIdx.x] += s2[threadIdx.x + off];
        }
        __syncthreads();
    }
    const float mean = s1[0] / (float)D;
    const float var  = s2[0] / (float)D - mean * mean;
    const float inv  = rsqrtf(var + EPS_);
    for (int i = threadIdx.x; i < D; i += 128)
        yr[i] = (xr[i] - mean) * inv * g[i] + b[i];
}

// ======================================================================
// attn[s,t] = softmax_t( scores[s,t] + G[s, s-t+S-1] )   (in place)
// G = Q_head @ Er^T (the Music-Transformer "skew" gather). Grid = S rows.
// ======================================================================
__global__ __launch_bounds__(256)
void rel_softmax_kernel(float* __restrict__ scores_head,
                        const float* __restrict__ G)
{
    const int s = blockIdx.x;
    float*       row = scores_head + (size_t)s * S_;
    const float* g   = G + (size_t)s * G_LD;
    __shared__ float red[256];

    float v[4];
    float lmax = -3.0e38f;
#pragma unroll
    for (int i = 0; i < 4; ++i) {
        const int t = threadIdx.x + (i << 8);
        v[i] = row[t] + g[s - t + (S_ - 1)];     // index in [s, s+1023] < 2047
        lmax = fmaxf(lmax, v[i]);
    }
    red[threadIdx.x] = lmax;
    __syncthreads();
    for (int off = 128; off > 0; off >>= 1) {
        if (threadIdx.x < off)
            red[threadIdx.x] = fmaxf(red[threadIdx.x], red[threadIdx.x + off]);
        __syncthreads();
    }
    const float mx = red[0];
    __syncthreads();

    float lsum = 0.0f;
#pragma unroll
    for (int i = 0; i < 4; ++i) { v[i] = expf(v[i] - mx); lsum += v[i]; }
    red[threadIdx.x] = lsum;
    __syncthreads();
    for (int off = 128; off > 0; off >>= 1) {
        if (threadIdx.x < off) red[threadIdx.x] += red[threadIdx.x + off];
        __syncthreads();
    }
    const float inv = 1.0f / red[0];
#pragma unroll
    for (int i = 0; i < 4; ++i)
        row[threadIdx.x + (i << 8)] = v[i] * inv;
}

// ======================================================================
// Host-side orchestration.
// Workspace layout (floats), total ~15.0 Mf = ~60 MiB:
//   hbuf,nbuf,qbuf,kbuf,vbuf,obuf : 6 * S*DM      = 3,145,728
//   f1buf (FFN hidden / final t1) : S*DFF         = 2,097,152
//   Gbuf  (skew, per head)        : S*G_LD        = 2,097,152
//   scores (all heads)            : NH*S*S        = 8,388,608
// ======================================================================
extern "C" void kernel_launch(void* const* d_in, const int* in_sizes, int n_in,
                              void* d_out, int out_size, void* d_ws, size_t ws_size,
                              hipStream_t stream)
{
    (void)in_sizes; (void)n_in; (void)out_size; (void)ws_size;

    const float* x     = (const float*)d_in[0];
    const float* embW  = (const float*)d_in[1];
    const float* embB  = (const float*)d_in[2];
    const float* Wq    = (const float*)d_in[3];
    const float* bq    = (const float*)d_in[4];
    const float* Wk    = (const float*)d_in[5];
    const float* bk    = (const float*)d_in[6];
    const float* Wv    = (const float*)d_in[7];
    const float* bv    = (const float*)d_in[8];
    const float* Wo    = (const float*)d_in[9];
    const float* bo    = (const float*)d_in[10];
    const float* Er    = (const float*)d_in[11];
    const float* ln1g  = (const float*)d_in[12];
    const float* ln1b  = (const float*)d_in[13];
    const float* W1    = (const float*)d_in[14];
    const float* b1    = (const float*)d_in[15];
    const float* W2    = (const float*)d_in[16];
    const float* b2    = (const float*)d_in[17];
    const float* ln2g  = (const float*)d_in[18];
    const float* ln2b  = (const float*)d_in[19];
    const float* lnfg  = (const float*)d_in[20];
    const float* lnfb  = (const float*)d_in[21];
    const float* Wi    = (const float*)d_in[22];
    const float* bi    = (const float*)d_in[23];
    const float* Wf    = (const float*)d_in[24];
    const float* bf    = (const float*)d_in[25];

    float* ws = (float*)d_ws;
    size_t off = 0;
    float* hbuf   = ws + off; off += (size_t)S_ * DM_;
    float* nbuf   = ws + off; off += (size_t)S_ * DM_;
    float* qbuf   = ws + off; off += (size_t)S_ * DM_;
    float* kbuf   = ws + off; off += (size_t)S_ * DM_;
    float* vbuf   = ws + off; off += (size_t)S_ * DM_;
    float* obuf   = ws + off; off += (size_t)S_ * DM_;
    float* f1buf  = ws + off; off += (size_t)S_ * DFF_;
    float* Gbuf   = ws + off; off += (size_t)S_ * G_LD;
    float* scores = ws + off; off += (size_t)NH_ * S_ * S_;
    float* t1buf  = f1buf;   // final-MLP hidden reuses FFN buffer (layers done)

    const dim3 blk(128);
    auto gemm_nn = [&](const float* A, int lda, const float* B, int ldb,
                       const float* bias, const float* R, int ldr,
                       float* C, int ldc, int M, int N, int K, int act) {
        dim3 grid((N + 127) / 128, M / 16);
        if ((N & 31) == 0)
            gemm_nn_kernel<false><<<grid, blk, 0, stream>>>(A, lda, B, ldb, bias, R, ldr, C, ldc, N, K, act);
        else
            gemm_nn_kernel<true ><<<grid, blk, 0, stream>>>(A, lda, B, ldb, bias, R, ldr, C, ldc, N, K, act);
    };
    auto gemm_nt = [&](const float* A, int lda, const float* B, int ldb,
                       float* C, int ldc, int M, int N, int K, float alpha) {
        dim3 grid((N + 127) / 128, M / 16);
        if ((N & 31) == 0)
            gemm_nt_kernel<false><<<grid, blk, 0, stream>>>(A, lda, B, ldb, C, ldc, N, K, alpha);
        else
            gemm_nt_kernel<true ><<<grid, blk, 0, stream>>>(A, lda, B, ldb, C, ldc, N, K, alpha);
    };

    const float scale = 0.125f;  // 1/sqrt(HD=64)

    // ---- embedding: h = x @ emb_W + emb_b ----
    gemm_nn(x, DIN_, embW, DM_, embB, nullptr, 0, hbuf, DM_, S_, DM_, DIN_, ACT_NONE);

    for (int l = 0; l < NL_; ++l) {
        const float* Wq_l = Wq + (size_t)l * DM_ * DM_;
        const float* Wk_l = Wk + (size_t)l * DM_ * DM_;
        const float* Wv_l = Wv + (size_t)l * DM_ * DM_;
        const float* Wo_l = Wo + (size_t)l * DM_ * DM_;
        const float* Er_l = Er + (size_t)l * ER_ROWS * HD_;
        const float* W1_l = W1 + (size_t)l * DM_ * DFF_;
        const float* W2_l = W2 + (size_t)l * DFF_ * DM_;

        // n = LN1(h)
        layernorm_kernel<<<S_, blk, 0, stream>>>(hbuf, ln1g + l * DM_, ln1b + l * DM_, nbuf, DM_);

        // q,k,v projections
        gemm_nn(nbuf, DM_, Wq_l, DM_, bq + l * DM_, nullptr, 0, qbuf, DM_, S_, DM_, DM_, ACT_NONE);
        gemm_nn(nbuf, DM_, Wk_l, DM_, bk + l * DM_, nullptr, 0, kbuf, DM_, S_, DM_, DM_, ACT_NONE);
        gemm_nn(nbuf, DM_, Wv_l, DM_, bv + l * DM_, nullptr, 0, vbuf, DM_, S_, DM_, DM_, ACT_NONE);

        // scores[h] = scale * Q_h K_h^T
        for (int h = 0; h < NH_; ++h)
            gemm_nt(qbuf + h * HD_, DM_, kbuf + h * HD_, DM_,
                    scores + (size_t)h * S_ * S_, S_, S_, S_, HD_, scale);

        // per head: G = Q_h Er^T, then skewed add + softmax (in place)
        for (int h = 0; h < NH_; ++h) {
            gemm_nt(qbuf + h * HD_, DM_, Er_l, HD_, Gbuf, G_LD, S_, ER_ROWS, HD_, 1.0f);
            rel_softmax_kernel<<<S_, dim3(256), 0, stream>>>(scores + (size_t)h * S_ * S_, Gbuf);
        }

        // o_h = attn_h @ V_h
        for (int h = 0; h < NH_; ++h)
            gemm_nn(scores + (size_t)h * S_ * S_, S_, vbuf + h * HD_, DM_,
                    nullptr, nullptr, 0, obuf + h * HD_, DM_, S_, HD_, S_, ACT_NONE);

        // h = h + o @ Wo + bo
        gemm_nn(obuf, DM_, Wo_l, DM_, bo + l * DM_, hbuf, DM_, hbuf, DM_, S_, DM_, DM_, ACT_NONE);

        // FFN: h = h + gelu(LN2(h) @ W1 + b1) @ W2 + b2
        layernorm_kernel<<<S_, blk, 0, stream>>>(hbuf, ln2g + l * DM_, ln2b + l * DM_, nbuf, DM_);
        gemm_nn(nbuf, DM_, W1_l, DFF_, b1 + l * DFF_, nullptr, 0, f1buf, DFF_, S_, DFF_, DM_, ACT_GELU);
        gemm_nn(f1buf, DFF_, W2_l, DM_, b2 + l * DM_, hbuf, DM_, hbuf, DM_, S_, DM_, DFF_, ACT_NONE);
    }

    // final: out = gelu(LNf(h) @ Wi + bi) @ Wf + bf
    layernorm_kernel<<<S_, blk, 0, stream>>>(hbuf, lnfg, lnfb, nbuf, DM_);
    gemm_nn(nbuf, DM_, Wi, 2 * DM_, bi, nullptr, 0, t1buf, 2 * DM_, S_, 2 * DM_, DM_, ACT_GELU);
    gemm_nn(t1buf, 2 * DM_, Wf, DOUT_, bf, nullptr, 0, (float*)d_out, DOUT_, S_, DOUT_, 2 * DM_, ACT_NONE);
}